// PhiNN_17033840296347
// MI455X (gfx1250) — compile-verified
//
#include <hip/hip_runtime.h>
#include <math.h>

typedef __attribute__((ext_vector_type(2))) float v2f;
typedef __attribute__((ext_vector_type(8))) float v8f;

#define KNDIM   2
#define KNCELLS 1000
#define KBATCH  4
#define KNSTEPS 251
#define KDT     0.001f
#define KSIGMA  0.001f
#define KXROW   (2 + KNCELLS * KNDIM + 5)   // 2007 floats per batch row

// D(16x16) = A(16x4) * B(4x16) + C, all f32, wave32.
__device__ __forceinline__ v8f wmma4(v2f a, v2f b, v8f c) {
  return __builtin_amdgcn_wmma_f32_16x16x4_f32(
      /*neg_a=*/false, a, /*neg_b=*/false, b,
      /*c_mod=*/(short)0, c, /*reuse_a=*/false, /*reuse_b=*/false);
}

// Hardware tanh (v_tanh_f32 is a CDNA5/gfx1250 TRANS op that co-executes with
// the matrix pipe). Fall back to a short v_exp_f32/v_rcp_f32 sequence.
__device__ __forceinline__ float fast_tanh(float x) {
#if __has_builtin(__builtin_amdgcn_tanhf)
  return __builtin_amdgcn_tanhf(x);
#elif __has_builtin(__builtin_amdgcn_tanh_f32)
  return __builtin_amdgcn_tanh_f32(x);
#else
  // tanh(x) = 1 - 2/(1 + 2^(x * 2*log2(e)))  (saturates correctly at +-inf)
  float e = __builtin_amdgcn_exp2f(x * 2.885390081777927f);
  return 1.0f - 2.0f * __builtin_amdgcn_rcpf(1.0f + e);
#endif
}

__global__ __launch_bounds__(32) void phi_sde_wmma(
    const float* __restrict__ x,   const float* __restrict__ dwg,
    const float* __restrict__ pw1, const float* __restrict__ pb1,
    const float* __restrict__ pw2, const float* __restrict__ pb2,
    const float* __restrict__ pw3, const float* __restrict__ pb3,
    const float* __restrict__ pw4, const float* __restrict__ pb4,
    const float* __restrict__ pw5,
    const float* __restrict__ twp, const float* __restrict__ tbp,
    float* __restrict__ out)
{
  // Weights/biases staged in LDS (row-major [out][in]); activation tiles
  // stored feature-major [feat][cell] for D-layout <-> B-layout exchange.
  __shared__ float sW1[32];    __shared__ float sB1[16];
  __shared__ float sW2[512];   __shared__ float sB2[32];
  __shared__ float sW3[1024];  __shared__ float sB3[32];
  __shared__ float sW4[512];   __shared__ float sB4[16];
  __shared__ float sW5[16];
  __shared__ float sH1[256];   // h1: 16 feats x 16 cells
  __shared__ float sH2[512];   // h2: 32 x 16
  __shared__ float sH3[512];   // h3: 32 x 16
  __shared__ float sT[512];    // backward d-tile exchange buffer

  const int lane = threadIdx.x;
  const int lm   = lane & 15;
  const int hi   = lane >> 4;          // 0 = lanes 0-15, 1 = lanes 16-31

  // Cooperative fill of LDS (blockDim == 32).
  if (lane < 32) sW1[lane] = pw1[lane];
  if (lane < 16) { sB1[lane] = pb1[lane]; sB4[lane] = pb4[lane]; sW5[lane] = pw5[lane]; }
  sB2[lane] = pb2[lane];
  sB3[lane] = pb3[lane];
  for (int i = lane; i < 512;  i += 32) { sW2[i] = pw2[i]; sW4[i] = pw4[i]; }
  for (int i = lane; i < 1024; i += 32) sW3[i] = pw3[i];

  // Per-lane point (16 cells per block; hi lanes mirror lo lanes harmlessly).
  const int p = blockIdx.x * 16 + lm;
  const int b = p / KNCELLS;
  const int c = p - b * KNCELLS;
  const float* xr = x + b * KXROW;
  const float t0    = xr[0];
  const float tcrit = xr[2 + KNCELLS * KNDIM + 0];
  const float p00   = xr[2 + KNCELLS * KNDIM + 1];
  const float p01   = xr[2 + KNCELLS * KNDIM + 2];
  const float p10   = xr[2 + KNCELLS * KNDIM + 3];
  const float p11   = xr[2 + KNCELLS * KNDIM + 4];
  float y0 = xr[2 + c * 2 + 0];
  float y1 = xr[2 + c * 2 + 1];
  const float tw00 = twp[0], tw01 = twp[1], tw10 = twp[2], tw11 = twp[3];
  const float tb0  = tbp[0], tb1  = tbp[1];

  __syncthreads();

  // Loop-invariant A fragments.
  // W1 as A (16x4, K cols 2..3 zeroed -> hi lanes zero).
  v2f aW1;
  aW1.x = hi ? 0.0f : sW1[lm * 2 + 0];
  aW1.y = hi ? 0.0f : sW1[lm * 2 + 1];
  // W1^T as A (16x16 padded: rows m>=2 zero).
  v2f aW1T[4];
#pragma unroll
  for (int kc = 0; kc < 4; ++kc) {
    int k0 = 4 * kc + 2 * hi;
    aW1T[kc].x = (lm < 2) ? sW1[k0 * 2 + lm]       : 0.0f;
    aW1T[kc].y = (lm < 2) ? sW1[(k0 + 1) * 2 + lm] : 0.0f;
  }
  // w5 broadcast matching C/D row layout (row = v + 8*hi).
  v8f w5C;
#pragma unroll
  for (int v = 0; v < 8; ++v) w5C[v] = sW5[v + 8 * hi];

#pragma unroll 1
  for (int s = 0; s < KNSTEPS; ++s) {
    // ---------------- forward L1: h1 = tanh(W1 y + b1) ----------------
    {
      v8f acc;
#pragma unroll
      for (int v = 0; v < 8; ++v) acc[v] = sB1[v + 8 * hi];
      v2f by; by.x = y0; by.y = y1;     // B rows 0,1 = y dims (hi-lane rows hit zero A cols)
      acc = wmma4(aW1, by, acc);
#pragma unroll
      for (int v = 0; v < 8; ++v) acc[v] = fast_tanh(acc[v]);
#pragma unroll
      for (int v = 0; v < 8; ++v) sH1[(v + 8 * hi) * 16 + lm] = acc[v];
    }
    __syncthreads();

    // ---------------- forward L2: h2 = tanh(W2 h1 + b2) ----------------
    {
      v2f bb[4];
#pragma unroll
      for (int kc = 0; kc < 4; ++kc) {
        int k0 = 4 * kc + 2 * hi;
        bb[kc].x = sH1[k0 * 16 + lm];
        bb[kc].y = sH1[(k0 + 1) * 16 + lm];
      }
#pragma unroll
      for (int t = 0; t < 2; ++t) {
        v8f acc;
#pragma unroll
        for (int v = 0; v < 8; ++v) acc[v] = sB2[16 * t + v + 8 * hi];
#pragma unroll
        for (int kc = 0; kc < 4; ++kc) {
          int k0 = 4 * kc + 2 * hi;
          v2f a;
          a.x = sW2[(16 * t + lm) * 16 + k0];
          a.y = sW2[(16 * t + lm) * 16 + k0 + 1];
          acc = wmma4(a, bb[kc], acc);
        }
#pragma unroll
        for (int v = 0; v < 8; ++v) acc[v] = fast_tanh(acc[v]);
#pragma unroll
        for (int v = 0; v < 8; ++v) sH2[(16 * t + v + 8 * hi) * 16 + lm] = acc[v];
      }
    }
    __syncthreads();

    // ---------------- forward L3: h3 = tanh(W3 h2 + b3) ----------------
    {
      v2f bb[8];
#pragma unroll
      for (int kc = 0; kc < 8; ++kc) {
        int k0 = 4 * kc + 2 * hi;
        bb[kc].x = sH2[k0 * 16 + lm];
        bb[kc].y = sH2[(k0 + 1) * 16 + lm];
      }
#pragma unroll
      for (int t = 0; t < 2; ++t) {
        v8f acc;
#pragma unroll
        for (int v = 0; v < 8; ++v) acc[v] = sB3[16 * t + v + 8 * hi];
#pragma unroll
        for (int kc = 0; kc < 8; ++kc) {
          int k0 = 4 * kc + 2 * hi;
          v2f a;
          a.x = sW3[(16 * t + lm) * 32 + k0];
          a.y = sW3[(16 * t + lm) * 32 + k0 + 1];
          acc = wmma4(a, bb[kc], acc);
        }
#pragma unroll
        for (int v = 0; v < 8; ++v) acc[v] = fast_tanh(acc[v]);
#pragma unroll
        for (int v = 0; v < 8; ++v) sH3[(16 * t + v + 8 * hi) * 16 + lm] = acc[v];
      }
    }
    __syncthreads();

    // ---------------- forward L4 + d4p = w5 * (1 - h4^2) ----------------
    v8f dcur;
    {
      v2f bb[8];
#pragma unroll
      for (int kc = 0; kc < 8; ++kc) {
        int k0 = 4 * kc + 2 * hi;
        bb[kc].x = sH3[k0 * 16 + lm];
        bb[kc].y = sH3[(k0 + 1) * 16 + lm];
      }
      v8f acc;
#pragma unroll
      for (int v = 0; v < 8; ++v) acc[v] = sB4[v + 8 * hi];
#pragma unroll
      for (int kc = 0; kc < 8; ++kc) {
        int k0 = 4 * kc + 2 * hi;
        v2f a;
        a.x = sW4[lm * 32 + k0];
        a.y = sW4[lm * 32 + k0 + 1];
        acc = wmma4(a, bb[kc], acc);
      }
#pragma unroll
      for (int v = 0; v < 8; ++v) {
        float h = fast_tanh(acc[v]);
        dcur[v] = w5C[v] * (1.0f - h * h);
      }
    }
#pragma unroll
    for (int v = 0; v < 8; ++v) sT[(v + 8 * hi) * 16 + lm] = dcur[v];
    __syncthreads();

    // ---------------- backward d3p = (1-h3^2) o (W4^T d4p) ----------------
    v8f d3a, d3b;
    {
      v2f bb[4];
#pragma unroll
      for (int kc = 0; kc < 4; ++kc) {
        int k0 = 4 * kc + 2 * hi;
        bb[kc].x = sT[k0 * 16 + lm];
        bb[kc].y = sT[(k0 + 1) * 16 + lm];
      }
#pragma unroll
      for (int t = 0; t < 2; ++t) {
        v8f acc = {};
#pragma unroll
        for (int kc = 0; kc < 4; ++kc) {
          int k0 = 4 * kc + 2 * hi;
          v2f a;
          a.x = sW4[k0 * 32 + 16 * t + lm];
          a.y = sW4[(k0 + 1) * 32 + 16 * t + lm];
          acc = wmma4(a, bb[kc], acc);
        }
#pragma unroll
        for (int v = 0; v < 8; ++v) {
          float h = sH3[(16 * t + v + 8 * hi) * 16 + lm];
          acc[v] *= (1.0f - h * h);
        }
        if (t == 0) d3a = acc; else d3b = acc;
      }
    }
    __syncthreads();
#pragma unroll
    for (int v = 0; v < 8; ++v) {
      sT[(v + 8 * hi) * 16 + lm]      = d3a[v];
      sT[(16 + v + 8 * hi) * 16 + lm] = d3b[v];
    }
    __syncthreads();

    // ---------------- backward d2p = (1-h2^2) o (W3^T d3p) ----------------
    v8f d2a, d2b;
    {
      v2f bb[8];
#pragma unroll
      for (int kc = 0; kc < 8; ++kc) {
        int k0 = 4 * kc + 2 * hi;
        bb[kc].x = sT[k0 * 16 + lm];
        bb[kc].y = sT[(k0 + 1) * 16 + lm];
      }
#pragma unroll
      for (int t = 0; t < 2; ++t) {
        v8f acc = {};
#pragma unroll
        for (int kc = 0; kc < 8; ++kc) {
          int k0 = 4 * kc + 2 * hi;
          v2f a;
          a.x = sW3[k0 * 32 + 16 * t + lm];
          a.y = sW3[(k0 + 1) * 32 + 16 * t + lm];
          acc = wmma4(a, bb[kc], acc);
        }
#pragma unroll
        for (int v = 0; v < 8; ++v) {
          float h = sH2[(16 * t + v + 8 * hi) * 16 + lm];
          acc[v] *= (1.0f - h * h);
        }
        if (t == 0) d2a = acc; else d2b = acc;
      }
    }
    __syncthreads();
#pragma unroll
    for (int v = 0; v < 8; ++v) {
      sT[(v + 8 * hi) * 16 + lm]      = d2a[v];
      sT[(16 + v + 8 * hi) * 16 + lm] = d2b[v];
    }
    __syncthreads();

    // ---------------- backward d1p = (1-h1^2) o (W2^T d2p) ----------------
    v8f d1;
    {
      v2f bb[8];
#pragma unroll
      for (int kc = 0; kc < 8; ++kc) {
        int k0 = 4 * kc + 2 * hi;
        bb[kc].x = sT[k0 * 16 + lm];
        bb[kc].y = sT[(k0 + 1) * 16 + lm];
      }
      v8f acc = {};
#pragma unroll
      for (int kc = 0; kc < 8; ++kc) {
        int k0 = 4 * kc + 2 * hi;
        v2f a;
        a.x = sW2[k0 * 16 + lm];
        a.y = sW2[(k0 + 1) * 16 + lm];
        acc = wmma4(a, bb[kc], acc);
      }
#pragma unroll
      for (int v = 0; v < 8; ++v) {
        float h = sH1[(v + 8 * hi) * 16 + lm];
        acc[v] *= (1.0f - h * h);
      }
      d1 = acc;
    }
    __syncthreads();
#pragma unroll
    for (int v = 0; v < 8; ++v) sT[(v + 8 * hi) * 16 + lm] = d1[v];
    __syncthreads();

    // ---------------- grad_y = W1^T d1p (rows 0,1 valid) ----------------
    float g0, g1;
    {
      v2f bb[4];
#pragma unroll
      for (int kc = 0; kc < 4; ++kc) {
        int k0 = 4 * kc + 2 * hi;
        bb[kc].x = sT[k0 * 16 + lm];
        bb[kc].y = sT[(k0 + 1) * 16 + lm];
      }
      v8f acc = {};
#pragma unroll
      for (int kc = 0; kc < 4; ++kc) acc = wmma4(aW1T[kc], bb[kc], acc);
      g0 = acc[0];   // M=0 row in lanes 0-15
      g1 = acc[1];   // M=1 row in lanes 0-15
    }

    // ---------------- tilt + Euler-Maruyama update ----------------
    float ts    = fmaf((float)s, KDT, t0);
    float sg0   = (ts < tcrit) ? p00 : p10;
    float sg1   = (ts < tcrit) ? p01 : p11;
    float tilt0 = fmaf(tw00, sg0, fmaf(tw01, sg1, tb0));
    float tilt1 = fmaf(tw10, sg0, fmaf(tw11, sg1, tb1));
    int dwi = ((b * KNSTEPS + s) * KNCELLS + c) * 2;
    float dw0 = dwg[dwi + 0];
    float dw1 = dwg[dwi + 1];
    y0 = fmaf(-KDT, g0 + tilt0, fmaf(KSIGMA, dw0, y0));
    y1 = fmaf(-KDT, g1 + tilt1, fmaf(KSIGMA, dw1, y1));
    __syncthreads();
  }

  if (lane < 16) {
    out[p * 2 + 0] = y0;
    out[p * 2 + 1] = y1;
  }
}

extern "C" void kernel_launch(void* const* d_in, const int* in_sizes, int n_in,
                              void* d_out, int out_size, void* d_ws, size_t ws_size,
                              hipStream_t stream) {
  (void)in_sizes; (void)n_in; (void)d_ws; (void)ws_size; (void)out_size;
  const float* x   = (const float*)d_in[0];
  const float* dw  = (const float*)d_in[1];
  const float* pw1 = (const float*)d_in[2];
  const float* pb1 = (const float*)d_in[3];
  const float* pw2 = (const float*)d_in[4];
  const float* pb2 = (const float*)d_in[5];
  const float* pw3 = (const float*)d_in[6];
  const float* pb3 = (const float*)d_in[7];
  const float* pw4 = (const float*)d_in[8];
  const float* pb4 = (const float*)d_in[9];
  const float* pw5 = (const float*)d_in[10];
  // d_in[11] = pb5: not needed for grad_phi
  const float* tw  = (const float*)d_in[12];
  const float* tb  = (const float*)d_in[13];

  dim3 grid((KBATCH * KNCELLS) / 16);   // 250 tiles of 16 cells
  dim3 block(32);                       // one wave32 per tile
  phi_sde_wmma<<<grid, block, 0, stream>>>(
      x, dw, pw1, pb1, pw2, pb2, pw3, pb3, pw4, pb4, pw5, tw, tb,
      (float*)d_out);
}